// FragmentEmbedder_1443109012245
// MI455X (gfx1250) — compile-verified
//
#include <hip/hip_runtime.h>
#include <stdint.h>

// FragmentEmbedder for MI455X (gfx1250).
//
// HBM floor ~68MB (~3us @23.3TB/s); dominant cost is 1.6GB of gene-gathered
// weight reads (weight1 = 4MB, L2-resident). WMMA does not apply (the 40x5
// weight differs per fragment -> no wave-shared B operand), so we use the
// CDNA5 Tensor Data Mover in GATHER mode: one tensor_load_to_lds per 16 lanes
// streams 16x800B weight rows (selected by 16-bit gene row-indices) into LDS,
// asynchronously (TENSORcnt), overlapped with the 40 hardware v_sin/v_cos ops
// each fragment needs. The matvec then reads LDS (ds_load_b128) instead of
// doing 50 random 16B L2 gathers per thread.
//
// Constraint rules learned from previous rounds:
//  - LDS pointer must ESCAPE into the asm ("v" operand) or LLVM proves
//    "no store to smem" and DCEs the whole compute tail.
//  - escaped (divergent) pointers must use "v", not "r" ("r" -> SGPR ->
//    illegal VGPR-to-SGPR copy).
//  - "s" descriptor tuple elements must be explicitly wave-uniform
//    (readfirstlane).
//  - do NOT launder the load pointer through an asm output: that kills the
//    addrspace(3) provenance and the matvec becomes FLAT loads instead of
//    ds_load_b128 (observed: ds=0, flat under vmem). Use an input-only
//    operand for ordering and index smem directly.

#define BLOCK     128          // 4 waves; 102.4KB LDS -> 3 blocks / WGP (320KB)
#define ROW_DW    200          // 40*5 floats per gene
#define ROW_BYTES 800
#define NG        5000

typedef uint32_t v4u __attribute__((ext_vector_type(4)));
typedef uint32_t v8u __attribute__((ext_vector_type(8)));

__global__ __launch_bounds__(BLOCK) void frag_embed_tdm(
    const float* __restrict__ coords,    // [N,2]
    const int*   __restrict__ gene_ix,   // [N]
    const float* __restrict__ weight1,   // [NG,40,5]
    const float* __restrict__ freqs,     // [20] (pairs of equal values)
    float*       __restrict__ out,       // [N,5]
    int n)
{
    __shared__ float smem[BLOCK * ROW_DW];           // 102400 bytes, offset 0

    const int tid  = blockIdx.x * BLOCK + (int)threadIdx.x;
    const int wave = (int)threadIdx.x >> 5;
    const int i    = (tid < n) ? tid : (n - 1);      // tail-safe clamp

    const int g = gene_ix[i];

    // ---------------- TDM gather descriptors (2 per wave) ----------------
    const uint64_t wbase = (uint64_t)(uintptr_t)weight1;
    // wave-uniform LDS byte offset, forced into an SGPR
    const uint32_t lds_wave =
        (uint32_t)__builtin_amdgcn_readfirstlane(wave * 32 * ROW_BYTES);

    // D# group 1 (256b): wg_mask=0, data_size=4B(code 2), no barrier/iterate/
    // pad, tensor_dim0=200, tensor_dim1=5000, tile_dim0=200, tile_dim1=16
    // indices, tensor_dim0_stride=200. dim1_stride/tile_dim2 ignored (gather).
    v8u g1;
    g1[0] = 0x00020000u;                 // data_size = 2 (4 bytes)
    g1[1] = (uint32_t)ROW_DW << 16;      // tensor_dim0[15:0]
    g1[2] = (uint32_t)NG << 16;          // tensor_dim0[31:16]=0 | tensor_dim1[15:0]
    g1[3] = (uint32_t)ROW_DW << 16;      // tensor_dim1[31:16]=0 | tile_dim0
    g1[4] = 16u;                         // tile_dim1 = #valid indices
    g1[5] = (uint32_t)ROW_DW;            // tensor_dim0_stride[31:0]
    g1[6] = 0u;
    g1[7] = 0u;

    // wave-private LDS destination (divergent generic pointer, escapes via "v")
    float* lds_dst = smem + (size_t)wave * 32 * ROW_DW;

#pragma unroll
    for (int d = 0; d < 2; ++d) {        // d=0: lanes 0-15, d=1: lanes 16-31
        v4u g0;
        g0[0] = 0x80000001u;             // gather_mode=1, 16-bit idx, count=1
        g0[1] = lds_wave + (uint32_t)(d * 16 * ROW_BYTES);   // lds_addr
        g0[2] = (uint32_t)wbase;                             // global_addr[31:0]
        g0[3] = (uint32_t)((wbase >> 32) & 0x01FFFFFFu)      // global_addr[56:32]
              | (2u << 30);                                  // type=2 (image)

        uint32_t h[16];
#pragma unroll
        for (int k = 0; k < 16; ++k)
            h[k] = (uint32_t)__builtin_amdgcn_readlane(g, d * 16 + k) & 0xFFFFu;

        v4u g2, g3;                      // 16x 16-bit row indices
        g2[0] = h[0]  | (h[1]  << 16);
        g2[1] = h[2]  | (h[3]  << 16);
        g2[2] = h[4]  | (h[5]  << 16);
        g2[3] = h[6]  | (h[7]  << 16);
        g3[0] = h[8]  | (h[9]  << 16);
        g3[1] = h[10] | (h[11] << 16);
        g3[2] = h[12] | (h[13] << 16);
        g3[3] = h[14] | (h[15] << 16);

        // %4 ("v") escapes the LDS array so AA must assume the asm writes it.
        asm volatile("tensor_load_to_lds %0, %1, %2, %3"
                     :: "s"(g0), "s"(g1), "s"(g2), "s"(g3),
                        "v"(lds_dst + (size_t)d * 16 * ROW_DW)
                     : "memory");
    }

    // ------------- sine/cos embedding, overlapped with the DMA -------------
    // shifts are {0, pi/2} pairs and each frequency repeats twice, so
    // emb[c*20+2q] = sin(coord_c*F_q), emb[c*20+2q+1] = cos(coord_c*F_q).
    const float cx = coords[2 * (size_t)i + 0];
    const float cy = coords[2 * (size_t)i + 1];
    float emb[40];
#pragma unroll
    for (int q = 0; q < 10; ++q) {
        const float F  = freqs[2 * q];
        const float ax = cx * F;
        const float ay = cy * F;
        emb[2 * q + 0]      = __sinf(ax);   // v_sin_f32
        emb[2 * q + 1]      = __cosf(ax);   // v_cos_f32
        emb[20 + 2 * q + 0] = __sinf(ay);
        emb[20 + 2 * q + 1] = __cosf(ay);
    }

    __builtin_amdgcn_s_wait_tensorcnt(0);   // weights resident in LDS

    // Ordering fence only: input-only operand keeps provenance of smem intact
    // (loads below still lower to ds_load_b128), memory clobber pins them
    // below the tensorcnt wait.
    asm volatile("" :: "v"(smem + (size_t)threadIdx.x * ROW_DW) : "memory");

    // ---------------- matvec from LDS: out[e] = sum_f emb[f]*w[f][e] --------
    const float4* wrow = (const float4*)(smem + (size_t)threadIdx.x * ROW_DW);
    float acc[5] = {0.f, 0.f, 0.f, 0.f, 0.f};
#pragma unroll
    for (int j = 0; j < 50; ++j) {
        const float4 w = wrow[j];           // ds_load_b128
        const int k = 4 * j;
        acc[(k + 0) % 5] += emb[(k + 0) / 5] * w.x;
        acc[(k + 1) % 5] += emb[(k + 1) / 5] * w.y;
        acc[(k + 2) % 5] += emb[(k + 2) / 5] * w.z;
        acc[(k + 3) % 5] += emb[(k + 3) / 5] * w.w;
    }

    if (tid < n) {
        float* o = out + (size_t)tid * 5;
#pragma unroll
        for (int e = 0; e < 5; ++e) {
            const float v = acc[e] > 0.f ? acc[e] : 0.f;   // ReLU
            __builtin_nontemporal_store(v, o + e);         // write-once stream
        }
    }
}

extern "C" void kernel_launch(void* const* d_in, const int* in_sizes, int n_in,
                              void* d_out, int out_size, void* d_ws, size_t ws_size,
                              hipStream_t stream) {
    const float* coords = (const float*)d_in[0];
    const int*   gene   = (const int*)  d_in[1];
    const float* w1     = (const float*)d_in[2];
    const float* freqs  = (const float*)d_in[3];
    // d_in[4] (shifts) folded analytically: {0, pi/2} pairs -> sin/cos.
    float* out = (float*)d_out;

    const int n      = in_sizes[0] / 2;          // [N,2] coordinates
    const int blocks = (n + BLOCK - 1) / BLOCK;

    hipLaunchKernelGGL(frag_embed_tdm, dim3(blocks), dim3(BLOCK), 0, stream,
                       coords, gene, w1, freqs, out, n);
}